// GDN_11441792876555
// MI455X (gfx1250) — compile-verified
//
#include <hip/hip_runtime.h>
#include <math.h>

typedef __attribute__((ext_vector_type(2))) float v2f;
typedef __attribute__((ext_vector_type(8))) float v8f;

#define Bn   128
#define Tn   128
#define Dn   256
#define DIM  64
#define KNN  20
#define NN   (Bn * Dn)        // 32768 nodes
#define NEGV (-1e30f)

// ---------------------------------------------------------------------------
// K1: cosine top-K over emb_w rows (stable ties: earlier index first).
// ---------------------------------------------------------------------------
__global__ void topk_kernel(const float* __restrict__ emb, int* __restrict__ topk) {
    __shared__ float norms[Dn];
    const int i = threadIdx.x;                 // one row per thread, 256 threads
    const float* wi = emb + i * DIM;
    float nrm = 0.f;
    for (int k = 0; k < DIM; ++k) nrm += wi[k] * wi[k];
    norms[i] = sqrtf(nrm);
    __syncthreads();

    float val[KNN];
    int   idx[KNN];
    for (int k = 0; k < KNN; ++k) { val[k] = -1e38f; idx[k] = 0; }
    const float inv_ni = 1.f / norms[i];
    for (int j = 0; j < Dn; ++j) {
        const float* wj = emb + j * DIM;
        float dt = 0.f;
        for (int k = 0; k < DIM; ++k) dt += wi[k] * wj[k];
        const float cv = dt * inv_ni / norms[j];
        if (cv > val[KNN - 1]) {
            int p = KNN - 1;
            while (p > 0 && val[p - 1] < cv) {   // strict: stable for ties
                val[p] = val[p - 1]; idx[p] = idx[p - 1]; --p;
            }
            val[p] = cv; idx[p] = j;
        }
    }
    for (int k = 0; k < KNN; ++k) topk[i * KNN + k] = idx[k];
}

// ---------------------------------------------------------------------------
// K2: xl = x @ lin_w  via V_WMMA_F32_16X16X4_F32.
//     x[n,t] = inputs[b*T*D + t*D + d], n = b*D + d.  One wave -> 16x64 tile.
// ---------------------------------------------------------------------------
__global__ void gemm_xl_kernel(const float* __restrict__ inputs,
                               const float* __restrict__ lin_w,
                               float* __restrict__ xl) {
    const int lane  = threadIdx.x & 31;
    const int wave  = threadIdx.x >> 5;
    const int gwave = blockIdx.x * (blockDim.x >> 5) + wave;
    const int m0    = gwave * 16;
    const int half  = lane >> 4;
    const int lc    = lane & 15;

    const int b  = m0 >> 8;
    const int d0 = m0 & 255;
    const float* xb = inputs + (size_t)b * (Tn * Dn) + d0;   // x[m0+r, t] = xb[t*Dn + r]

    v8f acc0 = {}, acc1 = {}, acc2 = {}, acc3 = {};
    for (int kk = 0; kk < Tn / 4; ++kk) {
        const int k0 = kk * 4;
        const int ka = k0 + 2 * half;
        v2f a;
        a.x = xb[(size_t)ka * Dn + lc];
        a.y = xb[(size_t)(ka + 1) * Dn + lc];
        const float* bw = lin_w + ka * DIM;                  // rows ka, ka+1
        v2f b0, b1, b2, b3;
        b0.x = bw[lc +  0]; b0.y = bw[DIM + lc +  0];
        b1.x = bw[lc + 16]; b1.y = bw[DIM + lc + 16];
        b2.x = bw[lc + 32]; b2.y = bw[DIM + lc + 32];
        b3.x = bw[lc + 48]; b3.y = bw[DIM + lc + 48];
        acc0 = __builtin_amdgcn_wmma_f32_16x16x4_f32(false, a, false, b0, (short)0, acc0, false, false);
        acc1 = __builtin_amdgcn_wmma_f32_16x16x4_f32(false, a, false, b1, (short)0, acc1, false, false);
        acc2 = __builtin_amdgcn_wmma_f32_16x16x4_f32(false, a, false, b2, (short)0, acc2, false, false);
        acc3 = __builtin_amdgcn_wmma_f32_16x16x4_f32(false, a, false, b3, (short)0, acc3, false, false);
    }
    // lane holds C[m0 + v + 8*half, ct*16 + lc] in acc{ct}[v]
    float* o = xl + (size_t)(m0 + 8 * half) * DIM + lc;
    for (int v = 0; v < 8; ++v) {
        o[(size_t)v * DIM +  0] = acc0[v];
        o[(size_t)v * DIM + 16] = acc1[v];
        o[(size_t)v * DIM + 32] = acc2[v];
        o[(size_t)v * DIM + 48] = acc3[v];
    }
}

// ---------------------------------------------------------------------------
// K3: attention logits si/sj per node.
// ---------------------------------------------------------------------------
__global__ void sisj_kernel(const float* __restrict__ xl, const float* __restrict__ emb,
                            const float* __restrict__ att_i, const float* __restrict__ att_j,
                            const float* __restrict__ att_em_i, const float* __restrict__ att_em_j,
                            float* __restrict__ si, float* __restrict__ sj) {
    const int n = blockIdx.x * blockDim.x + threadIdx.x;
    const float* xr = xl + (size_t)n * DIM;
    const float* er = emb + (size_t)(n & 255) * DIM;
    float a = 0.f, b = 0.f;
    for (int k = 0; k < DIM; ++k) {
        const float xv = xr[k], ev = er[k];
        a += xv * att_i[k] + ev * att_em_i[k];
        b += xv * att_j[k] + ev * att_em_j[k];
    }
    si[n] = a; sj[n] = b;
}

// ---------------------------------------------------------------------------
// K4: masked scatter-softmax over 20 kNN edges + self loop, then aggregation.
//     4 nodes per 256-thread block; thread = (local node, channel).
// ---------------------------------------------------------------------------
__global__ void attn_kernel(const float* __restrict__ xl,
                            const float* __restrict__ si, const float* __restrict__ sj,
                            const int* __restrict__ topk,
                            const float* __restrict__ gnn_bias,
                            float* __restrict__ agg) {
    __shared__ float alpha[4][KNN + 1];
    __shared__ int   srcs[4][KNN + 1];
    const int ln = threadIdx.x >> 6;
    const int ch = threadIdx.x & 63;
    const int n  = blockIdx.x * 4 + ln;
    const int d  = n & 255;
    const int bb = n >> 8;

    if (ch <= KNN) {
        const int sd   = (ch < KNN) ? topk[d * KNN + ch] : d;   // edge 20 = self loop
        const int src  = bb * 256 + sd;
        const bool valid = (ch == KNN) || (sd != d);            // remove self from kNN
        float sc = si[n] + sj[src];
        sc = (sc > 0.f) ? sc : 0.2f * sc;                       // leaky_relu 0.2
        alpha[ln][ch] = valid ? sc : NEGV;
        srcs[ln][ch]  = src;
    }
    __syncthreads();
    if (ch == 0) {
        float mx = NEGV;
        for (int e = 0; e <= KNN; ++e) mx = fmaxf(mx, alpha[ln][e]);
        float ex[KNN + 1];
        float den = 0.f;
        for (int e = 0; e <= KNN; ++e) {
            const float v = alpha[ln][e];
            const float t = (v <= -1e29f) ? 0.f : __expf(v - mx);
            ex[e] = t; den += t;
        }
        const float inv = 1.f / den;
        for (int e = 0; e <= KNN; ++e) alpha[ln][e] = ex[e] * inv;
    }
    __syncthreads();
    float acc = gnn_bias[ch];
    for (int e = 0; e <= KNN; ++e)
        acc += alpha[ln][e] * xl[(size_t)srcs[ln][e] * DIM + ch];
    agg[(size_t)n * DIM + ch] = acc;
}

// ---------------------------------------------------------------------------
// K5/K7: per-channel BN stats folded into scale/shift. One block per channel.
// ---------------------------------------------------------------------------
__global__ void chan_stats_kernel(const float* __restrict__ x,
                                  const float* __restrict__ g, const float* __restrict__ bt,
                                  float* __restrict__ scale, float* __restrict__ shift) {
    __shared__ float s1[256], s2[256];
    const int c = blockIdx.x;
    const int t = threadIdx.x;
    float a = 0.f, b = 0.f;
    for (int n = t; n < NN; n += 256) {
        const float v = x[(size_t)n * DIM + c];
        a += v; b += v * v;
    }
    s1[t] = a; s2[t] = b;
    __syncthreads();
    for (int o = 128; o > 0; o >>= 1) {
        if (t < o) { s1[t] += s1[t + o]; s2[t] += s2[t + o]; }
        __syncthreads();
    }
    if (t == 0) {
        const float mu  = s1[0] / (float)NN;
        const float var = s2[0] / (float)NN - mu * mu;
        const float sc  = g[c] * rsqrtf(var + 1e-5f);
        scale[c] = sc;
        shift[c] = bt[c] - mu * sc;
    }
}

// ---------------------------------------------------------------------------
// K6: gated = relu(bn1(agg)) * emb_w (broadcast over batch).
// ---------------------------------------------------------------------------
__global__ void gated_kernel(const float* __restrict__ agg,
                             const float* __restrict__ scale1, const float* __restrict__ shift1,
                             const float* __restrict__ emb, float* __restrict__ gated) {
    const size_t i = (size_t)blockIdx.x * blockDim.x + threadIdx.x;  // over N*DIM
    const int c = (int)(i & 63);
    const int n = (int)(i >> 6);
    const int d = n & 255;
    const float h = fmaxf(agg[i] * scale1[c] + shift1[c], 0.f);
    gated[i] = h * emb[d * DIM + c];
}

// ---------------------------------------------------------------------------
// K8: s = (relu(bn2(gated)) @ out_w1 + b1) @ out_w2 + b2  (WMMA + fused epilogue)
// ---------------------------------------------------------------------------
__global__ void gemm2_kernel(const float* __restrict__ gated,
                             const float* __restrict__ scale2, const float* __restrict__ shift2,
                             const float* __restrict__ out_w1, const float* __restrict__ out_b1,
                             const float* __restrict__ out_w2, const float* __restrict__ out_b2,
                             float* __restrict__ s) {
    const int lane  = threadIdx.x & 31;
    const int wave  = threadIdx.x >> 5;
    const int gwave = blockIdx.x * (blockDim.x >> 5) + wave;
    const int m0    = gwave * 16;
    const int half  = lane >> 4;
    const int lc    = lane & 15;

    const float* ar = gated + (size_t)(m0 + lc) * DIM;

    v8f acc0 = {}, acc1 = {}, acc2 = {}, acc3 = {};
    for (int kk = 0; kk < DIM / 4; ++kk) {
        const int k0 = kk * 4;
        const int ka = k0 + 2 * half;
        v2f a;
        a.x = fmaxf(ar[ka]     * scale2[ka]     + shift2[ka],     0.f);
        a.y = fmaxf(ar[ka + 1] * scale2[ka + 1] + shift2[ka + 1], 0.f);
        const float* bw = out_w1 + ka * DIM;
        v2f b0, b1, b2, b3;
        b0.x = bw[lc +  0]; b0.y = bw[DIM + lc +  0];
        b1.x = bw[lc + 16]; b1.y = bw[DIM + lc + 16];
        b2.x = bw[lc + 32]; b2.y = bw[DIM + lc + 32];
        b3.x = bw[lc + 48]; b3.y = bw[DIM + lc + 48];
        acc0 = __builtin_amdgcn_wmma_f32_16x16x4_f32(false, a, false, b0, (short)0, acc0, false, false);
        acc1 = __builtin_amdgcn_wmma_f32_16x16x4_f32(false, a, false, b1, (short)0, acc1, false, false);
        acc2 = __builtin_amdgcn_wmma_f32_16x16x4_f32(false, a, false, b2, (short)0, acc2, false, false);
        acc3 = __builtin_amdgcn_wmma_f32_16x16x4_f32(false, a, false, b3, (short)0, acc3, false, false);
    }
    // fused 64->1 projection: per-lane partials over this lane's 4 columns
    const float w2a = out_w2[lc +  0];
    const float w2b = out_w2[lc + 16];
    const float w2c = out_w2[lc + 32];
    const float w2d = out_w2[lc + 48];
    float p[8];
    for (int v = 0; v < 8; ++v)
        p[v] = acc0[v] * w2a + acc1[v] * w2b + acc2[v] * w2c + acc3[v] * w2d;
    // reduce across the 16 lanes of each half (rows v+8*half)
    for (int m = 1; m <= 8; m <<= 1)
        for (int v = 0; v < 8; ++v)
            p[v] += __shfl_xor(p[v], m, 32);
    if (lc == 0) {
        float cb = out_b2[0];
        for (int c = 0; c < DIM; ++c) cb += out_b1[c] * out_w2[c];
        for (int v = 0; v < 8; ++v)
            s[m0 + v + 8 * half] = p[v] + cb;
    }
}

// ---------------------------------------------------------------------------
// K9: bn3 stats (single channel over all N samples) -> scale/shift pair.
// ---------------------------------------------------------------------------
__global__ void scalar_stats_kernel(const float* __restrict__ s,
                                    const float* __restrict__ g3, const float* __restrict__ b3,
                                    float* __restrict__ sc_sh) {
    __shared__ float s1[1024], s2[1024];
    const int t = threadIdx.x;
    float a = 0.f, b = 0.f;
    for (int n = t; n < NN; n += 1024) { const float v = s[n]; a += v; b += v * v; }
    s1[t] = a; s2[t] = b;
    __syncthreads();
    for (int o = 512; o > 0; o >>= 1) {
        if (t < o) { s1[t] += s1[t + o]; s2[t] += s2[t + o]; }
        __syncthreads();
    }
    if (t == 0) {
        const float mu  = s1[0] / (float)NN;
        const float var = s2[0] / (float)NN - mu * mu;
        const float sc  = g3[0] * rsqrtf(var + 1e-5f);
        sc_sh[0] = sc;
        sc_sh[1] = b3[0] - mu * sc;
    }
}

// ---------------------------------------------------------------------------
// K10: out[b*D + d] = relu(bn3(s))
// ---------------------------------------------------------------------------
__global__ void final_kernel(const float* __restrict__ s, const float* __restrict__ sc_sh,
                             float* __restrict__ out) {
    const int n = blockIdx.x * blockDim.x + threadIdx.x;
    out[n] = fmaxf(s[n] * sc_sh[0] + sc_sh[1], 0.f);
}

// ---------------------------------------------------------------------------
extern "C" void kernel_launch(void* const* d_in, const int* in_sizes, int n_in,
                              void* d_out, int out_size, void* d_ws, size_t ws_size,
                              hipStream_t stream) {
    const float* inputs   = (const float*)d_in[0];
    const float* emb_w    = (const float*)d_in[1];
    const float* lin_w    = (const float*)d_in[2];
    const float* att_i    = (const float*)d_in[3];
    const float* att_j    = (const float*)d_in[4];
    const float* att_em_i = (const float*)d_in[5];
    const float* att_em_j = (const float*)d_in[6];
    const float* gnn_bias = (const float*)d_in[7];
    const float* bn1_g    = (const float*)d_in[8];
    const float* bn1_b    = (const float*)d_in[9];
    const float* bn2_g    = (const float*)d_in[10];
    const float* bn2_b    = (const float*)d_in[11];
    const float* out_w1   = (const float*)d_in[12];
    const float* out_b1   = (const float*)d_in[13];
    const float* out_w2   = (const float*)d_in[14];
    const float* out_b2   = (const float*)d_in[15];
    const float* bn3_g    = (const float*)d_in[16];
    const float* bn3_b    = (const float*)d_in[17];
    float* out = (float*)d_out;

    float* ws = (float*)d_ws;
    float* bufA   = ws;                         // xl, later reused for gated
    float* bufB   = bufA + (size_t)NN * DIM;    // agg
    float* si     = bufB + (size_t)NN * DIM;
    float* sj     = si + NN;
    float* svec   = sj + NN;
    float* scale1 = svec + NN;
    float* shift1 = scale1 + DIM;
    float* scale2 = shift1 + DIM;
    float* shift2 = scale2 + DIM;
    float* sc_sh3 = shift2 + DIM;
    int*   topk   = (int*)(sc_sh3 + 2);

    // 1. learned kNN graph (depends only on emb_w)
    topk_kernel<<<1, 256, 0, stream>>>(emb_w, topk);
    // 2. xl = x @ lin_w  (WMMA f32, 2048 waves, 4 waves/block)
    gemm_xl_kernel<<<NN / 16 / 4, 128, 0, stream>>>(inputs, lin_w, bufA);
    // 3. attention logits
    sisj_kernel<<<NN / 256, 256, 0, stream>>>(bufA, emb_w, att_i, att_j, att_em_i, att_em_j, si, sj);
    // 4. scatter-softmax + aggregation -> agg
    attn_kernel<<<NN / 4, 256, 0, stream>>>(bufA, si, sj, topk, gnn_bias, bufB);
    // 5. bn1 stats
    chan_stats_kernel<<<DIM, 256, 0, stream>>>(bufB, bn1_g, bn1_b, scale1, shift1);
    // 6. gated = relu(bn1(agg)) * emb_w  (xl dead -> reuse bufA)
    gated_kernel<<<(NN * DIM) / 256, 256, 0, stream>>>(bufB, scale1, shift1, emb_w, bufA);
    // 7. bn2 stats
    chan_stats_kernel<<<DIM, 256, 0, stream>>>(bufA, bn2_g, bn2_b, scale2, shift2);
    // 8. OutLayer: WMMA GEMM with bn2+relu folded in, 64->1 fused epilogue
    gemm2_kernel<<<NN / 16 / 4, 128, 0, stream>>>(bufA, scale2, shift2, out_w1, out_b1,
                                                  out_w2, out_b2, svec);
    // 9. bn3 stats
    scalar_stats_kernel<<<1, 1024, 0, stream>>>(svec, bn3_g, bn3_b, sc_sh3);
    // 10. final relu(bn3(.)) -> [B,1,D]
    final_kernel<<<NN / 256, 256, 0, stream>>>(svec, sc_sh3, out);
}